// DiagonalRydbergHamiltonian_15693810500005
// MI455X (gfx1250) — compile-verified
//
#include <hip/hip_runtime.h>

// DiagonalRydbergHamiltonian for n=22 qubits on gfx1250 (MI455X).
//
// out[b]      = halfRabi * sum_p psi[b ^ (1<<p)]  +  diag[b] * psi[b]
// diag[b]     = -detune*popcount(b) + sum_{p>q set} U[21-p][21-q]
//
// diag is built as a genuine f32 GEMM on the WMMA pipe:
//   b = (hi11, lo11);  diag = q_lo(lo) + [ P @ B ](hi, lo)
//   P (2048 x 12): P[hi][j] = sum_{i set in hi} U[10-i][21-j], P[hi][11] = q_hi(hi)
//   B (12 x 2048): B[j][lo] = bit_j(lo)  (j<11),  B[11][lo] = 1.0
// computed tile-wise with V_WMMA_F32_16X16X4_F32 (K=12 -> 3 k-steps).

#define N_Q   22
#define DIM   (1u << N_Q)          // 4194304
#define TBL   2048u                // 2^11

// workspace layout in floats
#define QLO_OFF   0u
#define P_OFF     2048u            // 2048 rows x 16 cols (K-padded, 8B-aligned v2f loads)
#define DIAG_OFF  65536u
#define WS_NEEDED ((size_t)(DIAG_OFF + DIM) * sizeof(float))

typedef __attribute__((ext_vector_type(2))) float v2f;
typedef __attribute__((ext_vector_type(8))) float v8f;

// ---------------------------------------------------------------------------
// Kernel 0: per-11-bit-pattern tables (2048 threads, trivial cost)
// ---------------------------------------------------------------------------
__global__ void build_tables(const float* __restrict__ U,
                             const float* __restrict__ detune,
                             float* __restrict__ ws) {
    unsigned t = blockIdx.x * blockDim.x + threadIdx.x;
    if (t >= TBL) return;
    float det = detune[0];

    // q_lo(t): bits p in [0,11)  (global positions 0..10)
    float qlo = 0.0f;
    for (int p = 0; p < 11; ++p) {
        if ((t >> p) & 1u) {
            qlo -= det;
            for (int q = 0; q < p; ++q)
                if ((t >> q) & 1u) qlo += U[(21 - p) * N_Q + (21 - q)];
        }
    }
    ws[QLO_OFF + t] = qlo;

    // q_hi(t) and cross row P[t][j]: hi bit i lives at global position 11+i
    float qhi = 0.0f;
    float prow[11];
    for (int j = 0; j < 11; ++j) prow[j] = 0.0f;
    for (int i = 0; i < 11; ++i) {
        if ((t >> i) & 1u) {
            qhi -= det;
            for (int i2 = 0; i2 < i; ++i2)
                if ((t >> i2) & 1u) qhi += U[(10 - i) * N_Q + (10 - i2)];
            for (int j = 0; j < 11; ++j)
                prow[j] += U[(10 - i) * N_Q + (21 - j)];   // pair (11+i, j)
        }
    }
    float* Pr = ws + P_OFF + t * 16;
    for (int j = 0; j < 11; ++j) Pr[j] = prow[j];
    Pr[11] = qhi;                                  // folded via B-row 11 == 1.0
    Pr[12] = 0.0f; Pr[13] = 0.0f; Pr[14] = 0.0f; Pr[15] = 0.0f;
}

// ---------------------------------------------------------------------------
// Kernel 1: diag table via WMMA f32 16x16x4. One wave per 16x16 output tile.
// 128x128 tiles = 16384 waves; grid/block chosen so EXEC is all-ones.
// A layout (ISA 7.12.2): lane=M%16; VGPR0/1 hold K={0,1} (lanes 0-15),
//   K={2,3} (lanes 16-31).  B mirrors with lane=N.  C/D: VGPR v holds row
//   M = v + 8*(lane>=16), col N = lane%16.
// ---------------------------------------------------------------------------
__global__ void build_diag_wmma(float* __restrict__ ws) {
    const float* __restrict__ P   = ws + P_OFF;
    const float* __restrict__ qlo = ws + QLO_OFF;
    float* __restrict__ diag      = ws + DIAG_OFF;

    unsigned wave  = (blockIdx.x * blockDim.x + threadIdx.x) >> 5;
    unsigned lane  = threadIdx.x & 31u;
    unsigned tileY = wave >> 7;          // hi tile  (0..127)
    unsigned tileX = wave & 127u;        // lo tile  (0..127)
    unsigned hi0   = tileY * 16u;
    unsigned lo0   = tileX * 16u;
    unsigned lmod  = lane & 15u;
    unsigned lhalf = lane >> 4;

    unsigned ncol = lo0 + lmod;          // N (low-11-bit value)
    unsigned mrow = hi0 + lmod;          // M for A-operand loads

    v8f acc = {};
#pragma unroll
    for (int k0 = 0; k0 < 12; k0 += 4) {
        // A: P[mrow][k0+2*lhalf .. +1]  (8B aligned: stride 16 floats, even col)
        v2f a = *(const v2f*)(P + mrow * 16u + (unsigned)k0 + 2u * lhalf);
        // B: bit matrix, K rows j0/j0+1; row 11 is the constant-1 row (q_hi)
        int j0 = k0 + 2 * (int)lhalf;
        v2f b;
        b.x = (j0 == 11)     ? 1.0f : (((ncol >> j0) & 1u)       ? 1.0f : 0.0f);
        b.y = (j0 + 1 == 11) ? 1.0f : (((ncol >> (j0 + 1)) & 1u) ? 1.0f : 0.0f);
        acc = __builtin_amdgcn_wmma_f32_16x16x4_f32(
                  false, a, false, b, (short)0, acc, false, false);
    }

    float ql = qlo[ncol];
#pragma unroll
    for (int v = 0; v < 8; ++v) {
        unsigned row = hi0 + (unsigned)v + 8u * lhalf;
        diag[row * TBL + ncol] = acc[v] + ql;   // coalesced 64B segments
    }
}

// ---------------------------------------------------------------------------
// Kernel 2: hot streaming kernel. 4 consecutive elements per thread (float4).
// Bit-0/1 flips are register shuffles; bits 2..21 are coalesced float4 loads
// (whole state fits in 192MB L2, so all neighbor traffic is L2-resident).
// ---------------------------------------------------------------------------
__global__ void __launch_bounds__(256)
apply_hamiltonian(const float* __restrict__ sre,
                  const float* __restrict__ sim,
                  const float* __restrict__ rabi,
                  const float* __restrict__ diag,
                  float* __restrict__ out) {
    unsigned tid  = blockIdx.x * blockDim.x + threadIdx.x;
    unsigned base = tid * 4u;
    float hr = 0.5f * rabi[0];

    float4 sr = *(const float4*)(sre + base);
    float4 si = *(const float4*)(sim + base);
    float4 dg = *(const float4*)(diag + base);

    // p = 0 flip (swap within pairs), p = 1 flip (swap pairs)
    float xr0 = sr.y + sr.z, xr1 = sr.x + sr.w, xr2 = sr.w + sr.x, xr3 = sr.z + sr.y;
    float xi0 = si.y + si.z, xi1 = si.x + si.w, xi2 = si.w + si.x, xi3 = si.z + si.y;

#pragma unroll
    for (int p = 2; p < N_Q; ++p) {
        unsigned nb = base ^ (1u << p);          // aligned: p >= 2
        float4 nr = *(const float4*)(sre + nb);
        float4 ni = *(const float4*)(sim + nb);
        xr0 += nr.x; xr1 += nr.y; xr2 += nr.z; xr3 += nr.w;
        xi0 += ni.x; xi1 += ni.y; xi2 += ni.z; xi3 += ni.w;
    }

    float4 orr, oii;
    orr.x = fmaf(hr, xr0, dg.x * sr.x);
    orr.y = fmaf(hr, xr1, dg.y * sr.y);
    orr.z = fmaf(hr, xr2, dg.z * sr.z);
    orr.w = fmaf(hr, xr3, dg.w * sr.w);
    oii.x = fmaf(hr, xi0, dg.x * si.x);
    oii.y = fmaf(hr, xi1, dg.y * si.y);
    oii.z = fmaf(hr, xi2, dg.z * si.z);
    oii.w = fmaf(hr, xi3, dg.w * si.w);

    *(float4*)(out + base)       = orr;
    *(float4*)(out + DIM + base) = oii;
}

// ---------------------------------------------------------------------------
// Fallback if workspace is too small: diag computed arithmetically per group
// of 4 elements (bits 0/1 contributions factored out of the pair loop).
// ---------------------------------------------------------------------------
__global__ void __launch_bounds__(256)
apply_hamiltonian_fallback(const float* __restrict__ sre,
                           const float* __restrict__ sim,
                           const float* __restrict__ rabi,
                           const float* __restrict__ detune,
                           const float* __restrict__ U,
                           float* __restrict__ out) {
    unsigned tid  = blockIdx.x * blockDim.x + threadIdx.x;
    unsigned base = tid * 4u;
    float hr  = 0.5f * rabi[0];
    float det = detune[0];

    float4 sr = *(const float4*)(sre + base);
    float4 si = *(const float4*)(sim + base);

    float xr0 = sr.y + sr.z, xr1 = sr.x + sr.w, xr2 = sr.w + sr.x, xr3 = sr.z + sr.y;
    float xi0 = si.y + si.z, xi1 = si.x + si.w, xi2 = si.w + si.x, xi3 = si.z + si.y;
#pragma unroll
    for (int p = 2; p < N_Q; ++p) {
        unsigned nb = base ^ (1u << p);
        float4 nr = *(const float4*)(sre + nb);
        float4 ni = *(const float4*)(sim + nb);
        xr0 += nr.x; xr1 += nr.y; xr2 += nr.z; xr3 += nr.w;
        xi0 += ni.x; xi1 += ni.y; xi2 += ni.z; xi3 += ni.w;
    }

    // diag for the 4-group: bits >=2 common part + bit0/bit1 coupling sums
    float common = 0.0f, s0 = 0.0f, s1 = 0.0f;
    for (int p = 2; p < N_Q; ++p) {
        if ((base >> p) & 1u) {
            common -= det;
            s0 += U[(21 - p) * N_Q + 21];   // pair (p, 0)
            s1 += U[(21 - p) * N_Q + 20];   // pair (p, 1)
            for (int q = 2; q < p; ++q)
                if ((base >> q) & 1u) common += U[(21 - p) * N_Q + (21 - q)];
        }
    }
    float u01 = U[20 * N_Q + 21];           // pair (1, 0)
    float d0 = common;
    float d1 = common + (s0 - det);
    float d2 = common + (s1 - det);
    float d3 = common + (s0 - det) + (s1 - det) + u01;

    float4 orr, oii;
    orr.x = fmaf(hr, xr0, d0 * sr.x);
    orr.y = fmaf(hr, xr1, d1 * sr.y);
    orr.z = fmaf(hr, xr2, d2 * sr.z);
    orr.w = fmaf(hr, xr3, d3 * sr.w);
    oii.x = fmaf(hr, xi0, d0 * si.x);
    oii.y = fmaf(hr, xi1, d1 * si.y);
    oii.z = fmaf(hr, xi2, d2 * si.z);
    oii.w = fmaf(hr, xi3, d3 * si.w);

    *(float4*)(out + base)       = orr;
    *(float4*)(out + DIM + base) = oii;
}

// ---------------------------------------------------------------------------
extern "C" void kernel_launch(void* const* d_in, const int* in_sizes, int n_in,
                              void* d_out, int out_size, void* d_ws, size_t ws_size,
                              hipStream_t stream) {
    (void)in_sizes; (void)n_in; (void)out_size;
    const float* sre    = (const float*)d_in[0];
    const float* sim    = (const float*)d_in[1];
    const float* rabi   = (const float*)d_in[2];
    const float* detune = (const float*)d_in[3];
    const float* U      = (const float*)d_in[4];
    float* out = (float*)d_out;
    float* ws  = (float*)d_ws;

    const unsigned apply_blocks = (DIM / 4u) / 256u;   // 4096

    if (ws_size >= WS_NEEDED) {
        build_tables<<<TBL / 256u, 256, 0, stream>>>(U, detune, ws);
        build_diag_wmma<<<(16384u * 32u) / 256u, 256, 0, stream>>>(ws);  // 2048 blocks
        apply_hamiltonian<<<apply_blocks, 256, 0, stream>>>(
            sre, sim, rabi, ws + DIAG_OFF, out);
    } else {
        apply_hamiltonian_fallback<<<apply_blocks, 256, 0, stream>>>(
            sre, sim, rabi, detune, U, out);
    }
}